// GCNSyntheticPerturbEdgeWeight_71476845740179
// MI455X (gfx1250) — compile-verified
//
#include <hip/hip_runtime.h>
#include <math.h>

#define NFEAT 512
#define NHID  256
#define NOUT  256
#define NCLS  8

typedef float v2f __attribute__((ext_vector_type(2)));
typedef float v8f __attribute__((ext_vector_type(8)));

// ---------------------------------------------------------------------------
// Utility: zero a float buffer
// ---------------------------------------------------------------------------
__global__ void zero_f32(float* __restrict__ p, int count) {
  int i = blockIdx.x * blockDim.x + threadIdx.x;
  if (i < count) p[i] = 0.0f;
}

// ---------------------------------------------------------------------------
// ew = sigmoid(params); weighted degree and count degree via fp32 atomics
// ---------------------------------------------------------------------------
__global__ void edge_deg(const float* __restrict__ prm, const int* __restrict__ dst,
                         float* __restrict__ ew, float* __restrict__ degw,
                         float* __restrict__ degc, int E) {
  int e = blockIdx.x * blockDim.x + threadIdx.x;
  if (e >= E) return;
  float w = 1.0f / (1.0f + expf(-prm[e]));
  ew[e] = w;
  atomicAdd(&degw[dst[e]], w);
  atomicAdd(&degc[dst[e]], 1.0f);
}

// dinv = rsqrt(deg + 1)  (self-loop weight 1); deg+1 >= 1 so always valid
__global__ void deg_finalize(const float* __restrict__ degw, const float* __restrict__ degc,
                             float* __restrict__ dinv, float* __restrict__ dinv3, int n) {
  int i = blockIdx.x * blockDim.x + threadIdx.x;
  if (i >= n) return;
  dinv[i]  = rsqrtf(degw[i] + 1.0f);
  dinv3[i] = rsqrtf(degc[i] + 1.0f);
}

// ---------------------------------------------------------------------------
// FP32 WMMA GEMM: C[M,N] = A[M,K] @ B[K,N], row-major.
// Requirements: M%16==0, N%64==0, K%4==0.
// One wave -> 16x64 output tile: one shared A fragment feeds four
// V_WMMA_F32_16X16X4_F32 per K-step (4 independent accumulator chains).
// Uniform scalar bases + 32-bit lane offsets -> saddr+voffset+imm loads.
// ---------------------------------------------------------------------------
__global__ __launch_bounds__(256) void gemm16x64_wmma_f32(
    const float* __restrict__ A, const float* __restrict__ B, float* __restrict__ C,
    int M, int N, int K) {
  const int lane = threadIdx.x & 31;
  const int wave = threadIdx.x >> 5;
  const int gw   = blockIdx.x * 8 + wave;
  const int tiles_n = N >> 6;                 // 64-wide col tiles
  const int tiles_m = M >> 4;                 // 16-tall row tiles
  if (gw >= tiles_m * tiles_n) return;        // wave-uniform -> EXEC stays ~0
  const int tm   = gw / tiles_n;
  const int tn   = gw % tiles_n;
  const int half = lane >> 4;                 // K-half select (ISA A/B layout)
  const int l15  = lane & 15;

  // 32-bit element offsets from uniform bases (max index < 2^26 here)
  const int aoff = (tm * 16 + l15) * K + 2 * half;   // A[row][k+2h], A[row][k+2h+1]
  const int boff = (2 * half) * N + tn * 64 + l15;   // B[k+2h][n], B[k+2h+1][n]

  v8f acc0 = {}, acc1 = {}, acc2 = {}, acc3 = {};
  #pragma unroll 4
  for (int k = 0; k < K; k += 4) {
    v2f a = *(const v2f*)(A + aoff + k);
    const float* bk = B + boff + k * N;
    v2f b0, b1, b2, b3;
    b0.x = bk[0];       b0.y = bk[N];
    b1.x = bk[16];      b1.y = bk[N + 16];
    b2.x = bk[32];      b2.y = bk[N + 32];
    b3.x = bk[48];      b3.y = bk[N + 48];
    acc0 = __builtin_amdgcn_wmma_f32_16x16x4_f32(false, a, false, b0, (short)0, acc0, false, false);
    acc1 = __builtin_amdgcn_wmma_f32_16x16x4_f32(false, a, false, b1, (short)0, acc1, false, false);
    acc2 = __builtin_amdgcn_wmma_f32_16x16x4_f32(false, a, false, b2, (short)0, acc2, false, false);
    acc3 = __builtin_amdgcn_wmma_f32_16x16x4_f32(false, a, false, b3, (short)0, acc3, false, false);
  }

  // C/D layout: VGPR v -> row 16*tm + v + 8*half, col = 16-group base + l15
  const int coff = (tm * 16 + 8 * half) * N + tn * 64 + l15;
  #pragma unroll
  for (int v = 0; v < 8; ++v) {
    float* cr = C + coff + v * N;
    cr[0]  = acc0[v];
    cr[16] = acc1[v];
    cr[32] = acc2[v];
    cr[48] = acc3[v];
  }
}

// ---------------------------------------------------------------------------
// Self-loop term: agg[i,f] = H[i,f] * dinv[i]^2
// ---------------------------------------------------------------------------
__global__ void selfloop_init(float* __restrict__ agg, const float* __restrict__ H,
                              const float* __restrict__ dinv, int n, int F) {
  int i = blockIdx.x * blockDim.x + threadIdx.x;
  if (i >= n * F) return;
  float d = dinv[i / F];
  agg[i] = H[i] * d * d;
}

// ---------------------------------------------------------------------------
// Edge scatter: agg[dst] += H[src] * (dinv[src]*w*dinv[dst]).
// Feature chunk (4 floats) varies fastest across threads -> coalesced gather
// of H rows and contiguous atomic targets (L2-resident fp32 atomics).
// ---------------------------------------------------------------------------
__global__ void edge_scatter(const float* __restrict__ H, float* __restrict__ agg,
                             const int* __restrict__ src, const int* __restrict__ dst,
                             const float* __restrict__ ew, const float* __restrict__ dinv,
                             int E, int F, int use_ew) {
  const int chunks = F >> 2;
  long long idx = (long long)blockIdx.x * blockDim.x + threadIdx.x;
  if (idx >= (long long)E * chunks) return;
  int e = (int)(idx / chunks);
  int c = ((int)(idx % chunks)) << 2;
  int s = src[e];
  int d = dst[e];
  float w   = use_ew ? ew[e] : 1.0f;
  float nrm = dinv[s] * w * dinv[d];
  const float4 h = *(const float4*)(H + (size_t)s * F + c);
  float* o = agg + (size_t)d * F + c;
  atomicAdd(o + 0, h.x * nrm);
  atomicAdd(o + 1, h.y * nrm);
  atomicAdd(o + 2, h.z * nrm);
  atomicAdd(o + 3, h.w * nrm);
}

// ---------------------------------------------------------------------------
// X = (relu?) (X + b)
// ---------------------------------------------------------------------------
__global__ void bias_act(float* __restrict__ X, const float* __restrict__ b,
                         int n, int F, int relu) {
  int i = blockIdx.x * blockDim.x + threadIdx.x;
  if (i >= n * F) return;
  float v = X[i] + b[i % F];
  X[i] = relu ? fmaxf(v, 0.0f) : v;
}

// ---------------------------------------------------------------------------
// Head: logits = [x1|x2|x3] @ Wlin + blin ; out = log_softmax(logits)
// One thread per node; Wlin (768x8, 24KB) is L2/L0 resident broadcast.
// ---------------------------------------------------------------------------
__global__ void final_head(const float* __restrict__ x1, const float* __restrict__ x2,
                           const float* __restrict__ x3, const float* __restrict__ Wl,
                           const float* __restrict__ bl, float* __restrict__ out, int n) {
  int i = blockIdx.x * blockDim.x + threadIdx.x;
  if (i >= n) return;
  float lg[NCLS];
  #pragma unroll
  for (int c = 0; c < NCLS; ++c) lg[c] = bl[c];

  const float* r1 = x1 + (size_t)i * NHID;
  for (int k = 0; k < NHID; ++k) {
    float v = r1[k];
    const float* w = Wl + (size_t)k * NCLS;
    #pragma unroll
    for (int c = 0; c < NCLS; ++c) lg[c] = fmaf(v, w[c], lg[c]);
  }
  const float* r2 = x2 + (size_t)i * NHID;
  for (int k = 0; k < NHID; ++k) {
    float v = r2[k];
    const float* w = Wl + (size_t)(NHID + k) * NCLS;
    #pragma unroll
    for (int c = 0; c < NCLS; ++c) lg[c] = fmaf(v, w[c], lg[c]);
  }
  const float* r3 = x3 + (size_t)i * NOUT;
  for (int k = 0; k < NOUT; ++k) {
    float v = r3[k];
    const float* w = Wl + (size_t)(2 * NHID + k) * NCLS;
    #pragma unroll
    for (int c = 0; c < NCLS; ++c) lg[c] = fmaf(v, w[c], lg[c]);
  }

  float mx = lg[0];
  #pragma unroll
  for (int c = 1; c < NCLS; ++c) mx = fmaxf(mx, lg[c]);
  float s = 0.0f;
  #pragma unroll
  for (int c = 0; c < NCLS; ++c) s += expf(lg[c] - mx);
  float lse = mx + logf(s);
  float* o = out + (size_t)i * NCLS;
  #pragma unroll
  for (int c = 0; c < NCLS; ++c) o[c] = lg[c] - lse;
}

// ---------------------------------------------------------------------------
extern "C" void kernel_launch(void* const* d_in, const int* in_sizes, int n_in,
                              void* d_out, int out_size, void* d_ws, size_t ws_size,
                              hipStream_t stream) {
  const float* x    = (const float*)d_in[0];
  const int*   ei   = (const int*)d_in[1];
  const float* ewp  = (const float*)d_in[2];
  const float* W1   = (const float*)d_in[3];
  const float* b1   = (const float*)d_in[4];
  const float* W2   = (const float*)d_in[5];
  const float* b2   = (const float*)d_in[6];
  const float* W3   = (const float*)d_in[7];
  const float* b3   = (const float*)d_in[8];
  const float* Wlin = (const float*)d_in[9];
  const float* blin = (const float*)d_in[10];

  const int n = in_sizes[0] / NFEAT;   // 50000 (multiple of 16)
  const int E = in_sizes[1] / 2;       // 400000
  const int* src = ei;
  const int* dst = ei + E;

  // workspace carve-up (floats)
  float* ws    = (float*)d_ws;
  float* ew    = ws;                   // E
  float* degw  = ew + E;               // n
  float* degc  = degw + n;             // n
  float* dinv  = degc + n;             // n
  float* dinv3 = dinv + n;             // n
  float* H     = dinv3 + n;            // n*256 GEMM output
  size_t nf    = (size_t)n * NHID;
  float* x1b   = H + nf;
  float* x2b   = x1b + nf;
  float* x3b   = x2b + nf;

  const dim3 B(256);

  // --- degree / edge-weight prep ---
  zero_f32<<<(2 * n + 255) / 256, B, 0, stream>>>(degw, 2 * n);  // degw,degc contiguous
  edge_deg<<<(E + 255) / 256, B, 0, stream>>>(ewp, dst, ew, degw, degc, E);
  deg_finalize<<<(n + 255) / 256, B, 0, stream>>>(degw, degc, dinv, dinv3, n);

  auto layer = [&](const float* Xin, float* Xout, const float* W, const float* bias,
                   const float* dv, int K, int F, int use_ew, int relu) {
    int tiles  = (n / 16) * (F / 64);
    int blocks = (tiles + 7) / 8;      // 8 waves per 256-thread block
    gemm16x64_wmma_f32<<<blocks, 256, 0, stream>>>(Xin, W, H, n, F, K);
    int tot = n * F;
    selfloop_init<<<(tot + 255) / 256, B, 0, stream>>>(Xout, H, dv, n, F);
    long long sc = (long long)E * (F / 4);
    edge_scatter<<<(int)((sc + 255) / 256), B, 0, stream>>>(H, Xout, src, dst, ew, dv, E, F, use_ew);
    bias_act<<<(tot + 255) / 256, B, 0, stream>>>(Xout, bias, n, F, relu);
  };

  layer(x,   x1b, W1, b1, dinv,  NFEAT, NHID, 1, 1);   // gc1 + relu
  layer(x1b, x2b, W2, b2, dinv,  NHID,  NHID, 1, 1);   // gc2 + relu
  layer(x2b, x3b, W3, b3, dinv3, NHID,  NOUT, 0, 0);   // gc3, unit edge weights

  final_head<<<(n + 255) / 256, B, 0, stream>>>(x1b, x2b, x3b, Wlin, blin,
                                                (float*)d_out, n);
}